// GATActorCritic_55001351193039
// MI455X (gfx1250) — compile-verified
//
#include <hip/hip_runtime.h>
#include <hip/hip_bf16.h>
#include <math.h>

typedef __attribute__((ext_vector_type(2))) float v2f;
typedef __attribute__((ext_vector_type(8))) float v8f;

#define HID_C 64
#define HEADS_C 4
#define IN_C 7
#define KCLUST 4

// ---------------------------------------------------------------------------
// f32 WMMA GEMM: C[M,NC] = A[M,KD] @ B[KD,NC], row-major. KD, NC, MT are
// compile-time so all strides fold to immediates / shifts; index math is
// 32-bit. One wave computes MT vertical 16x16 C tiles for one 16-col N tile,
// reusing each B fragment MT times. Launch geometry guarantees full tiles ->
// no predication, EXEC stays all-1s (WMMA requirement).
// A-layout (ISA 32-bit 16x4 f32): lanes 0-15 -> M=lane, VGPRs = K+0,K+1;
// lanes 16-31 -> M=lane-16, VGPRs = K+2,K+3. B mirrored (lane -> N column).
// C/D: VGPR r, lanes 0-15: M=r, N=lane; lanes 16-31: M=r+8, N=lane-16.
// ---------------------------------------------------------------------------
template <int KD, int MT, int NC>
__global__ void wmma_gemm_f32(const float* __restrict__ A,
                              const float* __restrict__ B,
                              float* __restrict__ C,
                              int mstart) {
  const int lane  = threadIdx.x & 31;
  const int wave  = threadIdx.x >> 5;
  const int half  = lane >> 4;        // 0: K+0/K+1, 1: K+2/K+3
  const int l     = lane & 15;
  const int n0    = (blockIdx.y * (blockDim.x >> 5) + wave) * 16;
  const int mbase = mstart + blockIdx.x * (16 * MT);

  v8f acc[MT];
#pragma unroll
  for (int t = 0; t < MT; ++t) acc[t] = (v8f){};

  const float* __restrict__ ap[MT];
#pragma unroll
  for (int t = 0; t < MT; ++t)
    ap[t] = A + (mbase + t * 16 + l) * KD;      // 32-bit index math

  const float* __restrict__ bp = B + n0 + l;

#pragma unroll 4
  for (int k = 0; k < KD; k += 4) {
    const int ko = k + 2 * half;
    const float* __restrict__ bb = bp + ko * NC;  // NC const -> shift/mad
    v2f b;
    b.x = bb[0];
    b.y = bb[NC];                                 // immediate offset load
#pragma unroll
    for (int t = 0; t < MT; ++t) {
      const float2 a2 = *(const float2*)(ap[t] + ko);   // aligned b64 load
      v2f a; a.x = a2.x; a.y = a2.y;
      acc[t] = __builtin_amdgcn_wmma_f32_16x16x4_f32(false, a, false, b,
                                                     (short)0, acc[t], false, false);
    }
  }

#pragma unroll
  for (int t = 0; t < MT; ++t) {
    float* __restrict__ cp = C + (mbase + t * 16 + 8 * half) * NC + n0 + l;
#pragma unroll
    for (int r = 0; r < 8; ++r)
      cp[r * NC] = acc[t][r];                     // immediate-offset stores
  }
}

// Pad x [N,7] -> xp [N,8] (col 7 zero) so the K-loop needs no guards
__global__ void pad_x_kernel(const float* __restrict__ x, float* __restrict__ xp, int n) {
  const int i = blockIdx.x * blockDim.x + threadIdx.x;
  if (i >= n * 8) return;
  const int r = i >> 3, c = i & 7;
  xp[i] = (c < IN_C) ? x[(size_t)r * IN_C + c] : 0.0f;
}

// Pad W1 [7,256] -> W1p [8,256] (row 7 zero)
__global__ void pad_w1_kernel(const float* __restrict__ W1, float* __restrict__ W1p) {
  const int i = blockIdx.x * blockDim.x + threadIdx.x;
  if (i >= 8 * 256) return;
  const int r = i >> 8, c = i & 255;
  W1p[i] = (r < IN_C) ? W1[r * 256 + c] : 0.0f;
}

// Per-head attention constants: ce[h] = dot(We1[h*64..], att_e1[h]); ce[4]=conv2
__global__ void ce_kernel(const float* __restrict__ We1, const float* __restrict__ ae1,
                          const float* __restrict__ We2, const float* __restrict__ ae2,
                          float* __restrict__ ce) {
  __shared__ float red[256];
  const int t = threadIdx.x;
  red[t] = We1[t] * ae1[t];
  __syncthreads();
  if (t < HEADS_C) {
    float s = 0.f;
    for (int c = 0; c < HID_C; ++c) s += red[t * HID_C + c];
    ce[t] = s;
  }
  if (t == HEADS_C) {
    float s = 0.f;
    for (int c = 0; c < HID_C; ++c) s += We2[c] * ae2[c];
    ce[HEADS_C] = s;
  }
}

// In-degree + sum of incoming edge attrs per dst node
__global__ void indeg_kernel(const int* __restrict__ ei, const float* __restrict__ ea,
                             int* __restrict__ deg, float* __restrict__ easum, int E) {
  const int e = blockIdx.x * blockDim.x + threadIdx.x;
  if (e >= E) return;
  const int d = ei[E + e];
  atomicAdd(&deg[d], 1);
  atomicAdd(&easum[d], ea[e]);
}

// loop_attr = mean incoming attr; degtot = indeg + 1 (self loop)
__global__ void loopattr_kernel(const int* __restrict__ deg, const float* __restrict__ easum,
                                float* __restrict__ loop_attr, int* __restrict__ degtot, int n) {
  const int i = blockIdx.x * blockDim.x + threadIdx.x;
  if (i >= n) return;
  const float c = (float)deg[i];
  loop_attr[i] = easum[i] / fmaxf(c, 1.0f);
  degtot[i] = deg[i] + 1;
}

// Single-block exclusive scan -> CSR offsets[0..n]
__global__ void scan_kernel(const int* __restrict__ deg, int* __restrict__ offs, int n) {
  __shared__ int buf[1024];
  __shared__ int carry;
  if (threadIdx.x == 0) { carry = 0; offs[0] = 0; }
  __syncthreads();
  for (int base = 0; base < n; base += 1024) {
    const int i = base + (int)threadIdx.x;
    buf[threadIdx.x] = (i < n) ? deg[i] : 0;
    __syncthreads();
    for (int o = 1; o < 1024; o <<= 1) {
      int t = (threadIdx.x >= (unsigned)o) ? buf[threadIdx.x - o] : 0;
      __syncthreads();
      buf[threadIdx.x] += t;
      __syncthreads();
    }
    if (i < n) offs[i + 1] = carry + buf[threadIdx.x];
    __syncthreads();
    if (threadIdx.x == 0) carry += buf[1023];
    __syncthreads();
  }
}

__global__ void csr_fill_edges(const int* __restrict__ ei, const float* __restrict__ ea,
                               const int* __restrict__ offs, int* __restrict__ cursor,
                               int* __restrict__ csr_src, float* __restrict__ csr_ea, int E) {
  const int e = blockIdx.x * blockDim.x + threadIdx.x;
  if (e >= E) return;
  const int d = ei[E + e];
  const int slot = offs[d] + atomicAdd(&cursor[d], 1);
  csr_src[slot] = ei[e];
  csr_ea[slot] = ea[e];
}

__global__ void csr_fill_loops(const int* __restrict__ offs, int* __restrict__ cursor,
                               const float* __restrict__ loop_attr,
                               int* __restrict__ csr_src, float* __restrict__ csr_ea, int n) {
  const int i = blockIdx.x * blockDim.x + threadIdx.x;
  if (i >= n) return;
  const int slot = offs[i] + atomicAdd(&cursor[i], 1);
  csr_src[slot] = i;
  csr_ea[slot] = loop_attr[i];
}

// al_src/al_dst per (node, head) for conv1
__global__ void dots1_kernel(const float* __restrict__ xw, const float* __restrict__ as_,
                             const float* __restrict__ ad_, float* __restrict__ al_s,
                             float* __restrict__ al_d, int n) {
  const int i = blockIdx.x * blockDim.x + threadIdx.x;
  if (i >= n * HEADS_C) return;
  const int node = i >> 2, h = i & 3;
  const float* __restrict__ row = xw + (size_t)node * (HEADS_C * HID_C) + h * HID_C;
  const float* __restrict__ vs = as_ + h * HID_C;
  const float* __restrict__ vd = ad_ + h * HID_C;
  float s = 0.f, d = 0.f;
#pragma unroll 8
  for (int c = 0; c < HID_C; ++c) { const float v = row[c]; s += v * vs[c]; d += v * vd[c]; }
  al_s[i] = s; al_d[i] = d;
}

__global__ void dots2_kernel(const float* __restrict__ xw, const float* __restrict__ as_,
                             const float* __restrict__ ad_, float* __restrict__ al_s,
                             float* __restrict__ al_d, int n) {
  const int i = blockIdx.x * blockDim.x + threadIdx.x;
  if (i >= n) return;
  const float* __restrict__ row = xw + (size_t)i * HID_C;
  float s = 0.f, d = 0.f;
#pragma unroll 8
  for (int c = 0; c < HID_C; ++c) { const float v = row[c]; s += v * as_[c]; d += v * ad_[c]; }
  al_s[i] = s; al_d[i] = d;
}

__device__ __forceinline__ float wave_max(float v) {
#pragma unroll
  for (int o = 16; o > 0; o >>= 1) v = fmaxf(v, __shfl_xor(v, o, 32));
  return v;
}
__device__ __forceinline__ float wave_sum(float v) {
#pragma unroll
  for (int o = 16; o > 0; o >>= 1) v += __shfl_xor(v, o, 32);
  return v;
}

// conv1: one wave32 per dst node. 4 heads x 64 ch; lane owns 8 channels.
__global__ void conv1_kernel(const float* __restrict__ xw1, const float* __restrict__ al_s,
                             const float* __restrict__ al_d, const float* __restrict__ ce,
                             const int* __restrict__ offs, const int* __restrict__ csr_src,
                             const float* __restrict__ csr_ea, float* __restrict__ ab,
                             const float* __restrict__ b1, float* __restrict__ h1, int n) {
  const int lane = threadIdx.x & 31;
  const int node = blockIdx.x * (blockDim.x >> 5) + (threadIdx.x >> 5);
  if (node >= n) return;
  const int beg = offs[node], end = offs[node + 1];
  const float ad0 = al_d[node * 4 + 0], ad1 = al_d[node * 4 + 1];
  const float ad2 = al_d[node * 4 + 2], ad3 = al_d[node * 4 + 3];
  const float ce0 = ce[0], ce1 = ce[1], ce2 = ce[2], ce3 = ce[3];

  float m0 = -3.4e38f, m1 = -3.4e38f, m2 = -3.4e38f, m3 = -3.4e38f;
  for (int s = beg + lane; s < end; s += 32) {
    const int src = csr_src[s];
    const float ea = csr_ea[s];
    const float* __restrict__ av = al_s + (size_t)src * 4;
    float a0 = av[0] + ad0 + ce0 * ea; a0 = a0 > 0.f ? a0 : 0.2f * a0;
    float a1 = av[1] + ad1 + ce1 * ea; a1 = a1 > 0.f ? a1 : 0.2f * a1;
    float a2 = av[2] + ad2 + ce2 * ea; a2 = a2 > 0.f ? a2 : 0.2f * a2;
    float a3 = av[3] + ad3 + ce3 * ea; a3 = a3 > 0.f ? a3 : 0.2f * a3;
    float* p = ab + (size_t)s * 4;
    p[0] = a0; p[1] = a1; p[2] = a2; p[3] = a3;
    m0 = fmaxf(m0, a0); m1 = fmaxf(m1, a1); m2 = fmaxf(m2, a2); m3 = fmaxf(m3, a3);
  }
  m0 = wave_max(m0); m1 = wave_max(m1); m2 = wave_max(m2); m3 = wave_max(m3);

  float s0 = 0.f, s1 = 0.f, s2 = 0.f, s3 = 0.f;
  for (int s = beg + lane; s < end; s += 32) {
    float* p = ab + (size_t)s * 4;
    const float e0 = __expf(p[0] - m0), e1 = __expf(p[1] - m1);
    const float e2 = __expf(p[2] - m2), e3 = __expf(p[3] - m3);
    p[0] = e0; p[1] = e1; p[2] = e2; p[3] = e3;
    s0 += e0; s1 += e1; s2 += e2; s3 += e3;
  }
  s0 = wave_sum(s0); s1 = wave_sum(s1); s2 = wave_sum(s2); s3 = wave_sum(s3);
  const float i0 = 1.f / (s0 + 1e-16f), i1 = 1.f / (s1 + 1e-16f);
  const float i2 = 1.f / (s2 + 1e-16f), i3 = 1.f / (s3 + 1e-16f);

  const int c0 = lane * 8;                 // channels [c0, c0+8)
  const int h  = c0 >> 6;                  // head for these channels
  const float invh = (h == 0) ? i0 : (h == 1) ? i1 : (h == 2) ? i2 : i3;
  float acc[8] = {0.f, 0.f, 0.f, 0.f, 0.f, 0.f, 0.f, 0.f};
  for (int s = beg; s < end; ++s) {
    const float w = ab[(size_t)s * 4 + h] * invh;
    const float4* __restrict__ p4 =
        (const float4*)(xw1 + (size_t)csr_src[s] * 256 + c0);
    const float4 u = p4[0], v = p4[1];
    acc[0] += w * u.x; acc[1] += w * u.y; acc[2] += w * u.z; acc[3] += w * u.w;
    acc[4] += w * v.x; acc[5] += w * v.y; acc[6] += w * v.z; acc[7] += w * v.w;
  }
  float* __restrict__ op = h1 + (size_t)node * 256 + c0;
#pragma unroll
  for (int j = 0; j < 8; ++j) {
    const float v = acc[j] + b1[c0 + j];
    op[j] = v > 0.f ? v : (__expf(v) - 1.f);   // ELU
  }
}

// conv2: one wave32 per dst node, single head, lane owns 2 channels
__global__ void conv2_kernel(const float* __restrict__ xw2, const float* __restrict__ al_s,
                             const float* __restrict__ al_d, const float* __restrict__ ce,
                             const int* __restrict__ offs, const int* __restrict__ csr_src,
                             const float* __restrict__ csr_ea, float* __restrict__ ab,
                             const float* __restrict__ b2, float* __restrict__ h2, int n) {
  const int lane = threadIdx.x & 31;
  const int node = blockIdx.x * (blockDim.x >> 5) + (threadIdx.x >> 5);
  if (node >= n) return;
  const int beg = offs[node], end = offs[node + 1];
  const float ad = al_d[node];
  const float cee = ce[HEADS_C];

  float m = -3.4e38f;
  for (int s = beg + lane; s < end; s += 32) {
    float a = al_s[csr_src[s]] + ad + cee * csr_ea[s];
    a = a > 0.f ? a : 0.2f * a;
    ab[s] = a;
    m = fmaxf(m, a);
  }
  m = wave_max(m);
  float sum = 0.f;
  for (int s = beg + lane; s < end; s += 32) {
    const float e = __expf(ab[s] - m);
    ab[s] = e;
    sum += e;
  }
  sum = wave_sum(sum);
  const float inv = 1.f / (sum + 1e-16f);

  const int c0 = lane * 2;
  float a0 = 0.f, a1 = 0.f;
  for (int s = beg; s < end; ++s) {
    const float w = ab[s] * inv;
    const float2 u = *(const float2*)(xw2 + (size_t)csr_src[s] * HID_C + c0);
    a0 += w * u.x; a1 += w * u.y;
  }
  h2[(size_t)node * HID_C + c0 + 0] = a0 + b2[c0 + 0];
  h2[(size_t)node * HID_C + c0 + 1] = a1 + b2[c0 + 1];
}

// Cluster pooling: LDS partials per block, one global atomicAdd per slot/block
__global__ void pool_kernel(const float* __restrict__ h2, const float* __restrict__ x,
                            const int* __restrict__ assign, float* __restrict__ zsum,
                            float* __restrict__ cfsum, float* __restrict__ cnt, int n) {
  __shared__ float zs[KCLUST * HID_C];
  __shared__ float cf[KCLUST];
  __shared__ float ct[KCLUST];
  const int t = threadIdx.x;
  if (t < KCLUST * HID_C) zs[t] = 0.f;
  if (t < KCLUST) { cf[t] = 0.f; ct[t] = 0.f; }
  __syncthreads();
  for (int i = blockIdx.x * blockDim.x + t; i < n; i += gridDim.x * blockDim.x) {
    const int a = assign[i];
    const float* __restrict__ row = h2 + (size_t)i * HID_C;
    float* zp = zs + a * HID_C;
    for (int c = 0; c < HID_C; ++c) atomicAdd(&zp[c], row[c]);
    atomicAdd(&cf[a], x[(size_t)i * IN_C + 6]);
    atomicAdd(&ct[a], 1.0f);
  }
  __syncthreads();
  if (t < KCLUST * HID_C) atomicAdd(&zsum[t], zs[t]);
  if (t < KCLUST) { atomicAdd(&cfsum[t], cf[t]); atomicAdd(&cnt[t], ct[t]); }
}

// Final head: masked means, MLP 65->64->1, softmax, write probs + z_flat
__global__ void head_kernel(const float* __restrict__ zsum, const float* __restrict__ cfsum,
                            const float* __restrict__ cnt, const float* __restrict__ A1,
                            const float* __restrict__ c1, const float* __restrict__ A2,
                            const float* __restrict__ c2, float* __restrict__ out) {
  __shared__ float z[KCLUST * HID_C];
  __shared__ float cf[KCLUST];
  __shared__ float hid[KCLUST * HID_C];
  __shared__ float logit[KCLUST];
  const int t = threadIdx.x;   // 64 threads
  for (int k = 0; k < KCLUST; ++k) {
    const float c = cnt[k];
    const float safe = fmaxf(c, 1.0f);
    float zv = zsum[k * HID_C + t] / safe;
    zv = (c > 0.f) ? zv : 0.0f;
    z[k * HID_C + t] = zv;
    out[4 + k * HID_C + t] = zv;           // z_flat
    if (t == 0) {
      const float v = cfsum[k] / safe;
      cf[k] = (c > 0.f) ? v : 0.0f;
    }
  }
  __syncthreads();
  for (int k = 0; k < KCLUST; ++k) {
    float s = c1[t];
    for (int i = 0; i < HID_C; ++i) s += z[k * HID_C + i] * A1[i * HID_C + t];
    s += cf[k] * A1[HID_C * HID_C + t];    // row 64 of A1 (the cf feature)
    hid[k * HID_C + t] = fmaxf(s, 0.0f);
  }
  __syncthreads();
  if (t < KCLUST) {
    float s = c2[0];
    for (int j = 0; j < HID_C; ++j) s += hid[t * HID_C + j] * A2[j];
    logit[t] = s;
  }
  __syncthreads();
  if (t == 0) {
    const float mx = fmaxf(fmaxf(logit[0], logit[1]), fmaxf(logit[2], logit[3]));
    const float e0 = __expf(logit[0] - mx), e1 = __expf(logit[1] - mx);
    const float e2 = __expf(logit[2] - mx), e3 = __expf(logit[3] - mx);
    const float den = e0 + e1 + e2 + e3;
    out[0] = e0 / den; out[1] = e1 / den; out[2] = e2 / den; out[3] = e3 / den;
  }
}

extern "C" void kernel_launch(void* const* d_in, const int* in_sizes, int n_in,
                              void* d_out, int out_size, void* d_ws, size_t ws_size,
                              hipStream_t stream) {
  const float* x        = (const float*)d_in[0];
  const int*   ei       = (const int*)d_in[1];
  const float* ea       = (const float*)d_in[2];
  const int*   assign   = (const int*)d_in[3];
  const float* W1       = (const float*)d_in[4];
  const float* att_src1 = (const float*)d_in[5];
  const float* att_dst1 = (const float*)d_in[6];
  const float* We1      = (const float*)d_in[7];
  const float* att_e1   = (const float*)d_in[8];
  const float* b1       = (const float*)d_in[9];
  const float* W2       = (const float*)d_in[10];
  const float* att_src2 = (const float*)d_in[11];
  const float* att_dst2 = (const float*)d_in[12];
  const float* We2      = (const float*)d_in[13];
  const float* att_e2   = (const float*)d_in[14];
  const float* b2       = (const float*)d_in[15];
  const float* A1       = (const float*)d_in[16];
  const float* c1       = (const float*)d_in[17];
  const float* A2       = (const float*)d_in[18];
  const float* c2       = (const float*)d_in[19];
  float* out = (float*)d_out;

  const int n  = in_sizes[0] / IN_C;   // 50000
  const int E  = in_sizes[1] / 2;      // 800000
  const int EP = E + n;                // edges + self loops

  // -------- workspace carve-out (256B aligned) --------
  char* w = (char*)d_ws;
  size_t off = 0;
  auto take = [&](size_t bytes) -> void* {
    void* p = w + off;
    off += (bytes + 255) & ~(size_t)255;
    return p;
  };
  float* xw1       = (float*)take((size_t)n * 256 * 4);
  float* h1        = (float*)take((size_t)n * 256 * 4);
  float* xw2       = (float*)take((size_t)n * HID_C * 4);
  float* h2        = (float*)take((size_t)n * HID_C * 4);
  float* alphabuf  = (float*)take((size_t)EP * 4 * 4);  // conv1: [EP,4]; conv2 reuses [EP]
  float* xp        = (float*)take((size_t)n * 8 * 4);   // x padded to K=8
  float* W1p       = (float*)take((size_t)8 * 256 * 4); // W1 padded to K=8
  float* al_s1     = (float*)take((size_t)n * 4 * 4);
  float* al_d1     = (float*)take((size_t)n * 4 * 4);
  float* al_s2     = (float*)take((size_t)n * 4);
  float* al_d2     = (float*)take((size_t)n * 4);
  float* easum     = (float*)take((size_t)n * 4);
  float* loop_attr = (float*)take((size_t)n * 4);
  int*   deg       = (int*)take((size_t)n * 4);
  int*   degtot    = (int*)take((size_t)n * 4);
  int*   offs      = (int*)take((size_t)(n + 1) * 4);
  int*   cursor    = (int*)take((size_t)n * 4);
  int*   csr_src   = (int*)take((size_t)EP * 4);
  float* csr_ea    = (float*)take((size_t)EP * 4);
  float* zsum      = (float*)take(KCLUST * HID_C * 4);
  float* cfsum     = (float*)take(KCLUST * 4);
  float* cnt       = (float*)take(KCLUST * 4);
  float* ce        = (float*)take(8 * 4);
  (void)ws_size; (void)n_in; (void)out_size;

  // -------- zero what we accumulate into --------
  hipMemsetAsync(deg,    0, (size_t)n * 4, stream);
  hipMemsetAsync(easum,  0, (size_t)n * 4, stream);
  hipMemsetAsync(cursor, 0, (size_t)n * 4, stream);
  hipMemsetAsync(zsum,   0, KCLUST * HID_C * 4, stream);
  hipMemsetAsync(cfsum,  0, KCLUST * 4, stream);
  hipMemsetAsync(cnt,    0, KCLUST * 4, stream);

  const int TB = 256;
  // graph structure + self loops
  ce_kernel<<<1, 256, 0, stream>>>(We1, att_e1, We2, att_e2, ce);
  indeg_kernel<<<(E + TB - 1) / TB, TB, 0, stream>>>(ei, ea, deg, easum, E);
  loopattr_kernel<<<(n + TB - 1) / TB, TB, 0, stream>>>(deg, easum, loop_attr, degtot, n);
  scan_kernel<<<1, 1024, 0, stream>>>(degtot, offs, n);
  csr_fill_edges<<<(E + TB - 1) / TB, TB, 0, stream>>>(ei, ea, offs, cursor, csr_src, csr_ea, E);
  csr_fill_loops<<<(n + TB - 1) / TB, TB, 0, stream>>>(offs, cursor, loop_attr, csr_src, csr_ea, n);

  // pad inputs so WMMA K-loops are guard-free
  pad_x_kernel<<<(n * 8 + TB - 1) / TB, TB, 0, stream>>>(x, xp, n);
  pad_w1_kernel<<<(8 * 256 + TB - 1) / TB, TB, 0, stream>>>(W1, W1p);

  // M = 50000 = mt64*64 + mrem16*16
  const int mt64   = n / 64;
  const int mrem16 = (n - mt64 * 64) / 16;

  // conv1: xw1 = xp @ W1p  (K=8, Nc=256; 16 N-tiles -> grid.y=4 with 4 waves/block)
  if (mt64 > 0)
    wmma_gemm_f32<8, 4, 256><<<dim3(mt64, 4), 128, 0, stream>>>(xp, W1p, xw1, 0);
  if (mrem16 > 0)
    wmma_gemm_f32<8, 1, 256><<<dim3(mrem16, 4), 128, 0, stream>>>(xp, W1p, xw1, mt64 * 64);

  dots1_kernel<<<(n * 4 + TB - 1) / TB, TB, 0, stream>>>(xw1, att_src1, att_dst1, al_s1, al_d1, n);
  conv1_kernel<<<(n + 7) / 8, 256, 0, stream>>>(xw1, al_s1, al_d1, ce, offs, csr_src, csr_ea,
                                                alphabuf, b1, h1, n);

  // conv2: xw2 = h1 @ W2  (K=256, Nc=64; 4 N-tiles -> grid.y=1 with 4 waves/block)
  if (mt64 > 0)
    wmma_gemm_f32<256, 4, 64><<<dim3(mt64, 1), 128, 0, stream>>>(h1, W2, xw2, 0);
  if (mrem16 > 0)
    wmma_gemm_f32<256, 1, 64><<<dim3(mrem16, 1), 128, 0, stream>>>(h1, W2, xw2, mt64 * 64);

  dots2_kernel<<<(n + TB - 1) / TB, TB, 0, stream>>>(xw2, att_src2, att_dst2, al_s2, al_d2, n);
  conv2_kernel<<<(n + 7) / 8, 256, 0, stream>>>(xw2, al_s2, al_d2, ce, offs, csr_src, csr_ea,
                                                alphabuf, b2, h2, n);

  // pooling + head
  pool_kernel<<<128, 256, 0, stream>>>(h2, x, assign, zsum, cfsum, cnt, n);
  head_kernel<<<1, 64, 0, stream>>>(zsum, cfsum, cnt, A1, c1, A2, c2, out);
}